// Sampler_2937757630765
// MI455X (gfx1250) — compile-verified
//
#include <hip/hip_runtime.h>
#include <hip/hip_bf16.h>
#include <stdint.h>

typedef __attribute__((ext_vector_type(2))) float v2f;
typedef __attribute__((ext_vector_type(8))) float v8f;
typedef int v4i __attribute__((vector_size(16)));

#define T_SEQ 4096
#define BATCH 64
#define DIM   2048
#define VOCAB 128000
#define EPS_SAMPLING 1e-5f

#if defined(__has_builtin)
#if __has_builtin(__builtin_amdgcn_global_load_async_to_lds_b128) && \
    __has_builtin(__builtin_amdgcn_s_wait_asynccnt)
#define HAVE_ASYNC_LDS 1
#endif
#endif

#ifdef HAVE_ASYNC_LDS
typedef __attribute__((address_space(3))) v4i v4i_lds;
#endif

// ---------------------------------------------------------------------------
// Kernel 1: gather last-token hidden rows, pre-scale by 1/temperature
// ---------------------------------------------------------------------------
__global__ void gather_scale_kernel(const float* __restrict__ hidden,
                                    const int* __restrict__ last_idx,
                                    const float* __restrict__ temps,
                                    float* __restrict__ hg) {
  int b = blockIdx.x;
  int row = last_idx[b];
  if (row < 0) row = 0;
  if (row >= T_SEQ) row = T_SEQ - 1;
  float t = temps[b];
  float scale = 1.0f / ((t < EPS_SAMPLING) ? 1.0f : t);
  const float4* src = (const float4*)(hidden + (size_t)row * DIM);
  float4* dst = (float4*)(hg + (size_t)b * DIM);
  for (int i = threadIdx.x; i < DIM / 4; i += blockDim.x) {
    float4 x = src[i];
    x.x *= scale; x.y *= scale; x.z *= scale; x.w *= scale;
    dst[i] = x;
  }
}

// ---------------------------------------------------------------------------
// Kernel 2: logits = hg[64,2048] @ emb[V,2048]^T via V_WMMA_F32_16X16X4_F32
// Block: 256 threads (8 wave32). Tile: M=64 (all), N=128 of V, K chunk 64.
// Double-buffered LDS; next chunk staged with GLOBAL_LOAD_ASYNC_TO_LDS_B128
// (ASYNCcnt) while WMMAs consume the current chunk.
// A frag (16x4 f32): lane l -> m = l%16, k = 2*(l/16)+j  (j = vgpr 0/1)
// B frag (4x16 f32): lane l -> n = l%16, k = 2*(l/16)+j
// C/D (16x16 f32):   vgpr r -> m = r + 8*(l/16), n = l%16
// ---------------------------------------------------------------------------
#define BK  64
#define LDP (BK + 4) // 68-float row pitch: 16B-aligned rows, conflict-free

__global__ __launch_bounds__(256)
void gemm_wmma_kernel(const float* __restrict__ hg,
                      const float* __restrict__ emb,
                      float* __restrict__ logits) {
  __shared__ float hT[2][BATCH * LDP];   // 2 x 64 x 68  (~34 KB)
  __shared__ float eT[2][128 * LDP];     // 2 x 128 x 68 (~68 KB)

  const int tid  = threadIdx.x;
  const int lane = tid & 31;
  const int w    = tid >> 5;          // wave id 0..7
  const int v0   = blockIdx.x * 128;  // base vocab column
  const int mr   = lane & 15;
  const int kb   = (lane >> 4) << 1;  // 0 or 2

  v8f acc[4] = {};

  // stage one K-chunk (64 wide) of h (64 rows) and E (128 rows) into buf
  auto stage = [&](int buf, int k0) {
#ifdef HAVE_ASYNC_LDS
#pragma unroll
    for (int i = 0; i < 4; ++i) {
      int idx = tid + 256 * i;
      int r = idx >> 4, c = (idx & 15) << 2;
      __builtin_amdgcn_global_load_async_to_lds_b128(
          (v4i*)(hg + (size_t)r * DIM + k0 + c),
          (v4i_lds*)&hT[buf][r * LDP + c], 0, 0);
    }
#pragma unroll
    for (int i = 0; i < 8; ++i) {
      int idx = tid + 256 * i;
      int r = idx >> 4, c = (idx & 15) << 2;
      __builtin_amdgcn_global_load_async_to_lds_b128(
          (v4i*)(emb + (size_t)(v0 + r) * DIM + k0 + c),
          (v4i_lds*)&eT[buf][r * LDP + c], 0, 0);
    }
#else
#pragma unroll
    for (int i = 0; i < 4; ++i) {
      int idx = tid + 256 * i;
      int r = idx >> 4, c = (idx & 15) << 2;
      *(float4*)&hT[buf][r * LDP + c] =
          *(const float4*)(hg + (size_t)r * DIM + k0 + c);
    }
#pragma unroll
    for (int i = 0; i < 8; ++i) {
      int idx = tid + 256 * i;
      int r = idx >> 4, c = (idx & 15) << 2;
      const float* p = emb + (size_t)(v0 + r) * DIM + k0 + c;
      *(float4*)&eT[buf][r * LDP + c] = *(const float4*)p;
      if (k0 + BK < DIM) __builtin_prefetch(p + BK, 0, 1); // global_prefetch_b8
    }
#endif
  };

  stage(0, 0);
#ifdef HAVE_ASYNC_LDS
  __builtin_amdgcn_s_wait_asynccnt(0);
#endif
  __syncthreads();

  int cur = 0;
  for (int k0 = 0; k0 < DIM; k0 += BK) {
    if (k0 + BK < DIM) stage(cur ^ 1, k0 + BK); // prefetch next chunk (async)

#pragma unroll 4
    for (int kk = 0; kk < BK; kk += 4) {
      v2f bfrag = *(const v2f*)&eT[cur][(16 * w + mr) * LDP + kk + kb];
#pragma unroll
      for (int mt = 0; mt < 4; ++mt) {
        v2f afrag = *(const v2f*)&hT[cur][(16 * mt + mr) * LDP + kk + kb];
        acc[mt] = __builtin_amdgcn_wmma_f32_16x16x4_f32(
            false, afrag, false, bfrag, (short)0, acc[mt], false, false);
      }
    }

#ifdef HAVE_ASYNC_LDS
    __builtin_amdgcn_s_wait_asynccnt(0);
#endif
    __syncthreads();
    cur ^= 1;
  }

  // ---- write back logits ----
#pragma unroll
  for (int mt = 0; mt < 4; ++mt) {
    int mbase = 16 * mt + ((lane >> 4) << 3);
    int v = v0 + 16 * w + mr;
#pragma unroll
    for (int r = 0; r < 8; ++r) {
      logits[(size_t)(mbase + r) * VOCAB + v] = acc[mt][r];
    }
  }
}

// ---------------------------------------------------------------------------
// Kernel 3: per-batch top-k / top-p / softmax / Gumbel sample (1 block/row)
// ---------------------------------------------------------------------------
#define ST 512

__device__ __forceinline__ uint32_t fkey(float x) {
  uint32_t u = __float_as_uint(x);
  return (u & 0x80000000u) ? ~u : (u | 0x80000000u); // monotone: bigger f -> bigger u
}

__device__ __forceinline__ float urand01(uint32_t b, uint32_t v) {
  uint32_t x = v * 0x9E3779B9u + b * 0x85EBCA6Bu + 0xC2B2AE35u;
  x ^= x >> 16; x *= 0x7FEB352Du; x ^= x >> 15; x *= 0x846CA68Bu; x ^= x >> 16;
  return (float)(x >> 8) * (1.0f / 16777216.0f) + 1e-20f; // (0,1)
}

__global__ __launch_bounds__(ST)
void sampler_kernel(const float* __restrict__ logits,
                    const float* __restrict__ temps,
                    const float* __restrict__ top_ps,
                    const int* __restrict__ top_ks,
                    float* __restrict__ out_tokens,
                    float* __restrict__ out_probs) {
  const int b = blockIdx.x;
  const int tid = threadIdx.x;
  const int lane = tid & 31;
  const int wv = tid >> 5; // 16 waves

  const float* row = logits + (size_t)b * VOCAB;
  float* prow = out_probs + (size_t)b * VOCAB;

  __shared__ uint32_t hist[256];
  __shared__ float fhist[256];
  __shared__ float sred[16];
  __shared__ int sredi[16];
  __shared__ uint32_t s_prefix;
  __shared__ int s_rem;
  __shared__ float s_lrem;
  __shared__ int s_flag;

  // ---- pass 1: max value + first argmax (greedy token, softmax shift) ----
  float mv = -3.4e38f; int mi = 0;
  for (int v = tid; v < VOCAB; v += ST) {
    float x = row[v];
    if (x > mv) { mv = x; mi = v; }
  }
  for (int o = 16; o > 0; o >>= 1) {
    float ov = __shfl_down(mv, o); int oi = __shfl_down(mi, o);
    if (ov > mv || (ov == mv && oi < mi)) { mv = ov; mi = oi; }
  }
  if (lane == 0) { sred[wv] = mv; sredi[wv] = mi; }
  __syncthreads();
  if (wv == 0) {
    mv = (lane < 16) ? sred[lane] : -3.4e38f;
    mi = (lane < 16) ? sredi[lane] : 0;
    for (int o = 8; o > 0; o >>= 1) {
      float ov = __shfl_down(mv, o); int oi = __shfl_down(mi, o);
      if (ov > mv || (ov == mv && oi < mi)) { mv = ov; mi = oi; }
    }
    if (lane == 0) { sred[0] = mv; sredi[0] = mi; }
  }
  __syncthreads();
  const float maxv = sred[0];
  const int maxIdx = sredi[0];
  __syncthreads();

  // ---- top-k: exact k-th largest via 4-pass radix select on fkey ----
  int k = top_ks[b];
  if (k < 1) k = 1;
  if (k > VOCAB) k = VOCAB;
  uint32_t prefix = 0; int remaining = k;
  for (int shift = 24; shift >= 0; shift -= 8) {
    const uint32_t himask = (shift == 24) ? 0u : (0xFFFFFFFFu << (shift + 8));
    for (int i = tid; i < 256; i += ST) hist[i] = 0;
    __syncthreads();
    for (int v = tid; v < VOCAB; v += ST) {
      uint32_t u = fkey(row[v]);
      if ((u & himask) == prefix) atomicAdd(&hist[(u >> shift) & 0xFF], 1u);
    }
    __syncthreads();
    if (tid == 0) {
      int cum = 0; uint32_t sel = 0; int rem = remaining;
      for (int bb = 255; bb >= 0; --bb) {
        int c = (int)hist[bb];
        if (cum + c >= remaining) { sel = (uint32_t)bb; rem = remaining - cum; break; }
        cum += c;
      }
      s_prefix = prefix | (sel << shift);
      s_rem = rem;
    }
    __syncthreads();
    prefix = s_prefix; remaining = s_rem;
    __syncthreads();
  }
  const uint32_t uK = prefix; // keep iff fkey >= uK

  // ---- Z over top-k set ----
  float zp = 0.f;
  for (int v = tid; v < VOCAB; v += ST) {
    float x = row[v];
    if (fkey(x) >= uK) zp += expf(x - maxv);
  }
  for (int o = 16; o > 0; o >>= 1) zp += __shfl_down(zp, o);
  if (lane == 0) sred[wv] = zp;
  __syncthreads();
  if (wv == 0) {
    float t = (lane < 16) ? sred[lane] : 0.f;
    for (int o = 8; o > 0; o >>= 1) t += __shfl_down(t, o);
    if (lane == 0) sred[0] = t;
  }
  __syncthreads();
  const float Z = sred[0];
  __syncthreads();

  // ---- top-p: weighted radix select over exp-mass.
  // kept-by-p = {x : S(x) <= top_p * Z}, S(x) = exp-mass of top-k elems >= x.
  // Boundary found at every level -> prefix is exact boundary VALUE,
  // kept = {u > uP} (strict). No boundary at some level -> whole bucket kept,
  // kept = {u >= pfx} (inclusive, low bits of pfx zero).
  const float tp = top_ps[b];
  const bool doP = (tp < 1.0f - EPS_SAMPLING);
  uint32_t uP = 0; bool strictP = false;
  if (doP) {
    float lrem = tp * Z;
    uint32_t pfx = 0;
    bool found_all = true;
    for (int shift = 24; shift >= 0; shift -= 8) {
      const uint32_t himask = (shift == 24) ? 0u : (0xFFFFFFFFu << (shift + 8));
      for (int i = tid; i < 256; i += ST) fhist[i] = 0.f;
      __syncthreads();
      for (int v = tid; v < VOCAB; v += ST) {
        float x = row[v];
        uint32_t u = fkey(x);
        if (u >= uK && (u & himask) == pfx)
          atomicAdd(&fhist[(u >> shift) & 0xFF], expf(x - maxv));
      }
      __syncthreads();
      if (tid == 0) {
        float acc = 0.f; int found = 0; uint32_t sel = 0; float lr = lrem;
        for (int bb = 255; bb >= 0; --bb) {
          float s = fhist[bb];
          if (acc + s > lrem) { sel = (uint32_t)bb; lr = lrem - acc; found = 1; break; }
          acc += s;
        }
        s_flag = found;
        s_prefix = pfx | (sel << shift);
        s_lrem = lr;
      }
      __syncthreads();
      if (!s_flag) { found_all = false; break; } // whole bucket kept
      pfx = s_prefix; lrem = s_lrem;
      __syncthreads();
    }
    uP = pfx;
    strictP = found_all;
  }
  __syncthreads();

  // ---- final kept set, Z', Gumbel argmax ----
  float z2 = 0.f; float gbest = -3.4e38f; int gidx = 0;
  for (int v = tid; v < VOCAB; v += ST) {
    float x = row[v];
    uint32_t u = fkey(x);
    if (u >= uK) {
      bool p_ok = !doP || (strictP ? (u > uP) : (u >= uP)) || (v == maxIdx);
      if (p_ok) {
        z2 += expf(x - maxv);
        float g = -logf(-logf(urand01((uint32_t)b, (uint32_t)v)));
        float sc = x + g;
        if (sc > gbest || (sc == gbest && v < gidx)) { gbest = sc; gidx = v; }
      }
    }
  }
  for (int o = 16; o > 0; o >>= 1) z2 += __shfl_down(z2, o);
  if (lane == 0) sred[wv] = z2;
  __syncthreads();
  if (wv == 0) {
    float t = (lane < 16) ? sred[lane] : 0.f;
    for (int o = 8; o > 0; o >>= 1) t += __shfl_down(t, o);
    if (lane == 0) sred[0] = t;
  }
  __syncthreads();
  const float Zp = sred[0];
  __syncthreads();

  for (int o = 16; o > 0; o >>= 1) {
    float ov = __shfl_down(gbest, o); int oi = __shfl_down(gidx, o);
    if (ov > gbest || (ov == gbest && oi < gidx)) { gbest = ov; gidx = oi; }
  }
  if (lane == 0) { sred[wv] = gbest; sredi[wv] = gidx; }
  __syncthreads();
  if (wv == 0) {
    gbest = (lane < 16) ? sred[lane] : -3.4e38f;
    gidx = (lane < 16) ? sredi[lane] : 0;
    for (int o = 8; o > 0; o >>= 1) {
      float ov = __shfl_down(gbest, o); int oi = __shfl_down(gidx, o);
      if (ov > gbest || (ov == gbest && oi < gidx)) { gbest = ov; gidx = oi; }
    }
    if (lane == 0) sredi[0] = gidx;
  }
  __syncthreads();
  const int sampIdx = sredi[0];

  // ---- write probs ----
  const float invZ = (Zp > 0.f) ? (1.0f / Zp) : 0.f;
  for (int v = tid; v < VOCAB; v += ST) {
    float x = row[v];
    uint32_t u = fkey(x);
    float p = 0.f;
    if (u >= uK) {
      bool p_ok = !doP || (strictP ? (u > uP) : (u >= uP)) || (v == maxIdx);
      if (p_ok) p = expf(x - maxv) * invZ;
    }
    prow[v] = p;
  }

  if (tid == 0) {
    float t = temps[b];
    int tok = (t < EPS_SAMPLING) ? maxIdx : sampIdx;
    if (tok < 0 || tok >= VOCAB) tok = 0;
    out_tokens[b] = (float)tok;
  }
}

// ---------------------------------------------------------------------------
extern "C" void kernel_launch(void* const* d_in, const int* in_sizes, int n_in,
                              void* d_out, int out_size, void* d_ws, size_t ws_size,
                              hipStream_t stream) {
  const float* hidden = (const float*)d_in[0];
  const float* emb    = (const float*)d_in[1];
  const int*   lasti  = (const int*)d_in[2];
  const float* temps  = (const float*)d_in[3];
  const float* topps  = (const float*)d_in[4];
  const int*   topks  = (const int*)d_in[5];

  float* out_tokens = (float*)d_out;              // B ids (as float)
  float* out_probs  = (float*)d_out + BATCH;      // B x V probs

  float* logits = (float*)d_ws;                           // 64 x 128000 f32
  float* hg     = logits + (size_t)BATCH * VOCAB;         // 64 x 2048 f32

  gather_scale_kernel<<<BATCH, 256, 0, stream>>>(hidden, lasti, temps, hg);
  gemm_wmma_kernel<<<VOCAB / 128, 256, 0, stream>>>(hg, emb, logits);
  sampler_kernel<<<BATCH, ST, 0, stream>>>(logits, temps, topps, topks,
                                           out_tokens, out_probs);
}